// IndexKernelSingle_32238024524408
// MI455X (gfx1250) — compile-verified
//
#include <hip/hip_runtime.h>

// out[i] = <covar_factor[x[i]], covar_factor[y[i]]> + (x[i]==y[i]) * std[x[i]]^2
//
// Strategy: one wave32 handles a tile of 16 pairs. A = 16 gathered x-rows
// (16x32 f32), B = 16 gathered y-rows as columns (32x16 f32). Chain eight
// V_WMMA_F32_16X16X4_F32 ops over K=32; the diagonal of the 16x16 f32
// accumulator is the 16 desired dot products. Memory traffic is identical to
// a scalar gather-dot (each row fetched once per tile, all L2-resident since
// the whole table is 2 MB << 192 MB L2), so the wasted off-diagonal FLOPs are
// free and we ride the CDNA5 matrix pipe.

typedef float v2f __attribute__((ext_vector_type(2)));
typedef float v8f __attribute__((ext_vector_type(8)));

#define RANK 32
#define PAIRS_PER_WAVE 16
#define WAVES_PER_BLOCK 8
#define BLOCK_THREADS (WAVES_PER_BLOCK * 32)

__global__ __launch_bounds__(BLOCK_THREADS) void covar_pair_gather_wmma(
    const int* __restrict__ xIdx,
    const int* __restrict__ yIdx,
    const float* __restrict__ stdv,
    const float* __restrict__ cf,
    float* __restrict__ out,
    int n_pairs)
{
    const int lane = threadIdx.x & 31;
    const int wave = threadIdx.x >> 5;
    const int tile = blockIdx.x * WAVES_PER_BLOCK + wave;
    const int base = tile * PAIRS_PER_WAVE;
    if (base >= n_pairs) return;            // wave-uniform: EXEC stays all-1s

    const int col = lane & 15;              // which pair this lane serves
    const int hi  = lane >> 4;              // 0 -> K sub-pair {0,1}; 1 -> {2,3}

    const int xi = xIdx[base + col];
    const int yi = yIdx[base + col];

    // Per-lane fragment base: row start + 8B half selected by `hi`.
    const float* arow = cf + (size_t)xi * RANK + hi * 2;
    const float* brow = cf + (size_t)yi * RANK + hi * 2;

    v8f c = {};
#pragma unroll
    for (int k = 0; k < 8; ++k) {
        // A frag (16x4 f32): lanes 0-15 give row col, K={4k,4k+1};
        //                    lanes 16-31 give row col, K={4k+2,4k+3}.
        // B frag (4x16 f32): same striping with y-rows as columns.
        v2f a = *(const v2f*)(arow + 4 * k);
        v2f b = *(const v2f*)(brow + 4 * k);
        // 8 args: (neg_a, A, neg_b, B, c_mod, C, reuse_a, reuse_b)
        c = __builtin_amdgcn_wmma_f32_16x16x4_f32(
                false, a, false, b, (short)0, c, false, false);
    }

    // Diagonal extraction. Element (p,p) of D:
    //   p < 8 : lane p,     VGPR p
    //   p >= 8: lane p+16,  VGPR p-8
    // => storing lanes: 0-7 and 24-31; pair = lane&15; vgpr = lane&7.
    const int v = lane & 7;
    float d = c[0];
    d = (v == 1) ? c[1] : d;
    d = (v == 2) ? c[2] : d;
    d = (v == 3) ? c[3] : d;
    d = (v == 4) ? c[4] : d;
    d = (v == 5) ? c[5] : d;
    d = (v == 6) ? c[6] : d;
    d = (v == 7) ? c[7] : d;

    const bool storing = (lane < 8) || (lane >= 24);
    if (storing) {
        float val = d;
        if (xi == yi) {
            float s = stdv[xi];
            val += s * s;
        }
        out[base + col] = val;
    }
}

extern "C" void kernel_launch(void* const* d_in, const int* in_sizes, int n_in,
                              void* d_out, int out_size, void* d_ws, size_t ws_size,
                              hipStream_t stream) {
    const int*   x    = (const int*)d_in[0];
    const int*   y    = (const int*)d_in[1];
    const float* stdv = (const float*)d_in[2];
    const float* cf   = (const float*)d_in[3];
    float*       out  = (float*)d_out;

    const int n_pairs = in_sizes[0];                       // 4194304
    const int pairs_per_block = PAIRS_PER_WAVE * WAVES_PER_BLOCK;  // 128
    const int blocks = (n_pairs + pairs_per_block - 1) / pairs_per_block;

    covar_pair_gather_wmma<<<blocks, BLOCK_THREADS, 0, stream>>>(
        x, y, stdv, cf, out, n_pairs);
}